// ChannelLinearAttention_14431090115170
// MI455X (gfx1250) — compile-verified
//
#include <hip/hip_runtime.h>
#include <hip/hip_bf16.h>

// Problem constants (fixed by the reference): x is (16, 64, 32, 32, 32) fp32
#define BB 16
#define CC 64
#define NN 32768          // 32*32*32
#define EPSI 1e-6f

typedef __attribute__((ext_vector_type(2))) float v2f;
typedef __attribute__((ext_vector_type(8))) float v8f;

// -------- Workspace layout (in floats) --------
//  w   : [BB*NN]        inverse column L2 norm
//  vs  : [BB*NN]        column sum over C
//  mat : [BB*64*64]     Gram matrix (atomically accumulated, memset to 0)
//  tail: [BB*64]        tailor_sum
#define W_OFF    0
#define VS_OFF   (BB * NN)
#define MAT_OFF  (2 * BB * NN)
#define TAIL_OFF (2 * BB * NN + BB * 64 * 64)

// ============================================================
// Kernel 1: per-(b,n) column stats: inv-norm and sum over C
// ============================================================
__global__ void k_colstats(const float* __restrict__ x,
                           float* __restrict__ w, float* __restrict__ vs) {
    int idx = blockIdx.x * 256 + threadIdx.x;      // over BB*NN
    int b = idx / NN;
    int n = idx - b * NN;
    const float* xp = x + (size_t)b * CC * NN + n;
    float s = 0.f, ss = 0.f;
#pragma unroll
    for (int c = 0; c < CC; ++c) {
        float v = xp[(size_t)c * NN];
        s += v;
        ss += v * v;
    }
    w[idx]  = 1.0f / sqrtf(ss);
    vs[idx] = s;
}

// ============================================================
// Kernel 2: Gram matrix mat[b] += X_tile * diag(w) * X_tile^T
// grid = BB * KCH blocks, 512 threads (16 waves), each wave owns
// one 16x16 tile of the 64x64 output; fp32 WMMA 16x16x4.
// Software-pipelined global->LDS staging + dual accumulators.
// ============================================================
#define KCH 32                         // K-chunks per batch
#define KCOLS (NN / KCH)               // 1024 columns per block
#define LDSW 68                        // padded LDS stride (dwords)

__global__ void k_gram(const float* __restrict__ x,
                       const float* __restrict__ w,
                       float* __restrict__ mat) {
    __shared__ __align__(16) float xs[64 * LDSW];  // raw tile   [row c][col k]
    __shared__ __align__(16) float xq[64 * LDSW];  // w-scaled   [row m][col k]

    int b     = blockIdx.x >> 5;       // /KCH
    int chunk = blockIdx.x & (KCH - 1);
    int kbase = chunk * KCOLS;

    int tid  = threadIdx.x;
    int lane = tid & 31;
    int wid  = tid >> 5;               // 0..15
    int ti   = wid >> 2;               // output row tile (c)
    int tj   = wid & 3;                // output col tile (m)
    int lm   = lane & 15;
    int koff = (lane >> 4) * 2;

    const float* xb = x + (size_t)b * CC * NN;
    const float* wb = w + (size_t)b * NN;

    // per-thread staging registers for one 64x64 tile (8 elems each)
    int srow[8], scol[8];
#pragma unroll
    for (int r = 0; r < 8; ++r) {
        int e   = tid + r * 512;
        srow[r] = e >> 6;
        scol[r] = e & 63;
    }

    float rv[8], rw[8];
#pragma unroll
    for (int r = 0; r < 8; ++r) {                     // prefetch first tile
        rv[r] = xb[(size_t)srow[r] * NN + (kbase + scol[r])];
        rw[r] = wb[kbase + scol[r]];
    }

    v8f acc0 = {0.f, 0.f, 0.f, 0.f, 0.f, 0.f, 0.f, 0.f};
    v8f acc1 = {0.f, 0.f, 0.f, 0.f, 0.f, 0.f, 0.f, 0.f};

    int aoff = (ti * 16 + lm) * LDSW + koff;
    int boff = (tj * 16 + lm) * LDSW + koff;

    for (int k0 = kbase; k0 < kbase + KCOLS; k0 += 64) {
        // commit staged registers to LDS
#pragma unroll
        for (int r = 0; r < 8; ++r) {
            xs[srow[r] * LDSW + scol[r]] = rv[r];
            xq[srow[r] * LDSW + scol[r]] = rv[r] * rw[r];
        }
        __syncthreads();
        // kick off next tile's global loads while WMMAs run
        if (k0 + 64 < kbase + KCOLS) {
#pragma unroll
            for (int r = 0; r < 8; ++r) {
                rv[r] = xb[(size_t)srow[r] * NN + (k0 + 64 + scol[r])];
                rw[r] = wb[k0 + 64 + scol[r]];
            }
        }
        // 16 WMMAs, two independent accumulator chains
#pragma unroll
        for (int kk = 0; kk < 64; kk += 8) {
            v2f a0 = *(const v2f*)&xs[aoff + kk];
            v2f b0 = *(const v2f*)&xq[boff + kk];
            v2f a1 = *(const v2f*)&xs[aoff + kk + 4];
            v2f b1 = *(const v2f*)&xq[boff + kk + 4];
            acc0 = __builtin_amdgcn_wmma_f32_16x16x4_f32(
                false, a0, false, b0, (short)0, acc0, false, false);
            acc1 = __builtin_amdgcn_wmma_f32_16x16x4_f32(
                false, a1, false, b1, (short)0, acc1, false, false);
        }
        __syncthreads();
    }

    // accumulate partial 16x16 tile into mat[b]
    float* mp = mat + (size_t)b * 64 * 64;
    int cbase = ti * 16 + (lane >> 4) * 8;
    int m     = tj * 16 + lm;
#pragma unroll
    for (int r = 0; r < 8; ++r)
        atomicAdd(&mp[(cbase + r) * 64 + m], acc0[r] + acc1[r]);
}

// ============================================================
// Kernel 3: tailor[b,c] = 1/(N + sum_n x[b,c,n]*(vs*w+eps)*w)
// one block per (b,c)
// ============================================================
__global__ void k_tailor(const float* __restrict__ x,
                         const float* __restrict__ w,
                         const float* __restrict__ vs,
                         float* __restrict__ tail) {
    __shared__ float red[256];
    int b = blockIdx.x >> 6;
    int c = blockIdx.x & 63;
    const float* xp = x + ((size_t)b * CC + c) * NN;
    const float* wb = w + (size_t)b * NN;
    const float* vb = vs + (size_t)b * NN;

    float acc = 0.f;
    for (int n = threadIdx.x; n < NN; n += 256) {
        float wn = wb[n];
        float t  = (vb[n] * wn + EPSI) * wn;
        acc += xp[n] * t;
    }
    red[threadIdx.x] = acc;
    __syncthreads();
    for (int s = 128; s > 0; s >>= 1) {
        if (threadIdx.x < s) red[threadIdx.x] += red[threadIdx.x + s];
        __syncthreads();
    }
    if (threadIdx.x == 0)
        tail[blockIdx.x] = 1.0f / ((float)NN + red[0]);
}

// ============================================================
// Kernel 4: out = x + gamma*tailor[c]*(vs[n] + mat[b] @ (x*w))
// grid = BB * (NN/32), 256 threads (8 waves: 4 row-tiles x 2 col-tiles)
// ============================================================
#define QSW 66                          // padded stride for transposed q slab

__global__ void k_apply(const float* __restrict__ x,
                        const float* __restrict__ gamma,
                        const float* __restrict__ w,
                        const float* __restrict__ vs,
                        const float* __restrict__ mat,
                        const float* __restrict__ tail,
                        float* __restrict__ out) {
    __shared__ __align__(16) float ms[64 * LDSW];  // mat[b]: [c][m]
    __shared__ __align__(16) float qs[32 * QSW];   // transposed: [ncol][m]

    int b     = blockIdx.x >> 10;       // / (NN/32)
    int chunk = blockIdx.x & 1023;
    int n0    = chunk * 32;

    int tid  = threadIdx.x;
    int lane = tid & 31;
    int wid  = tid >> 5;                // 0..7
    int ti   = wid >> 1;                // c tile 0..3
    int tj   = wid & 1;                 // n tile 0..1
    int lm   = lane & 15;
    int koff = (lane >> 4) * 2;

    const float* xb = x + (size_t)b * CC * NN;
    const float* wb = w + (size_t)b * NN;
    const float* mb = mat + (size_t)b * 64 * 64;

    // load mat[b] (64x64) into LDS
#pragma unroll
    for (int r = 0; r < 16; ++r) {
        int e   = tid + r * 256;
        int row = e >> 6;
        int col = e & 63;
        ms[row * LDSW + col] = mb[e];
    }
    // load q slab (64 rows m x 32 cols n), scaled by w, store transposed
#pragma unroll
    for (int r = 0; r < 8; ++r) {
        int e   = tid + r * 256;
        int m   = e >> 5;
        int col = e & 31;
        float v = xb[(size_t)m * NN + (n0 + col)] * wb[n0 + col];
        qs[col * QSW + m] = v;
    }
    __syncthreads();

    v8f acc0 = {0.f, 0.f, 0.f, 0.f, 0.f, 0.f, 0.f, 0.f};
    v8f acc1 = {0.f, 0.f, 0.f, 0.f, 0.f, 0.f, 0.f, 0.f};
    int aoff = (ti * 16 + lm) * LDSW + koff;
    int boff = (tj * 16 + lm) * QSW + koff;
#pragma unroll
    for (int kk = 0; kk < 64; kk += 8) {
        v2f a0 = *(const v2f*)&ms[aoff + kk];
        v2f b0 = *(const v2f*)&qs[boff + kk];
        v2f a1 = *(const v2f*)&ms[aoff + kk + 4];
        v2f b1 = *(const v2f*)&qs[boff + kk + 4];
        acc0 = __builtin_amdgcn_wmma_f32_16x16x4_f32(
            false, a0, false, b0, (short)0, acc0, false, false);
        acc1 = __builtin_amdgcn_wmma_f32_16x16x4_f32(
            false, a1, false, b1, (short)0, acc1, false, false);
    }

    // fused epilogue: out = x + g * tail[c] * (vs[n] + acc)
    float g   = gamma[0];
    int nn    = n0 + tj * 16 + lm;
    float vsn = vs[(size_t)b * NN + nn];
    int cbase = ti * 16 + (lane >> 4) * 8;
    const float* tb = tail + b * 64;
#pragma unroll
    for (int r = 0; r < 8; ++r) {
        int c = cbase + r;
        size_t idx = ((size_t)b * CC + c) * NN + nn;
        out[idx] = x[idx] + g * tb[c] * (vsn + acc0[r] + acc1[r]);
    }
}

// ============================================================
extern "C" void kernel_launch(void* const* d_in, const int* in_sizes, int n_in,
                              void* d_out, int out_size, void* d_ws, size_t ws_size,
                              hipStream_t stream) {
    const float* x     = (const float*)d_in[0];
    const float* gamma = (const float*)d_in[1];
    float* out = (float*)d_out;
    float* ws  = (float*)d_ws;

    float* w    = ws + W_OFF;
    float* vs   = ws + VS_OFF;
    float* mat  = ws + MAT_OFF;
    float* tail = ws + TAIL_OFF;

    // zero the Gram accumulator (capture-legal async memset)
    hipMemsetAsync(mat, 0, (size_t)BB * 64 * 64 * sizeof(float), stream);

    k_colstats<<<(BB * NN) / 256, 256, 0, stream>>>(x, w, vs);
    k_gram<<<BB * KCH, 512, 0, stream>>>(x, w, mat);
    k_tailor<<<BB * CC, 256, 0, stream>>>(x, w, vs, tail);
    k_apply<<<BB * (NN / 32), 256, 0, stream>>>(x, gamma, w, vs, mat, tail, out);
}